// ModulatedPlasticRNNCell_403726926395
// MI455X (gfx1250) — compile-verified
//
#include <hip/hip_runtime.h>

// ---------------------------------------------------------------------------
// ModulatedPlasticRNNCell on MI455X (gfx1250)
// B=64, I=512, H=1024.  hebb is 256MB f32 -> problem is HBM-streaming bound:
//   pass1 read 256MB + pass2 read 256MB + write 256MB ~ 768MB ~ 33us @23.3TB/s
// GEMMs (x@fc_w.T, h_pre@weight) are ~0.2 GFLOP: done exactly in f32 with
// V_WMMA_F32_16X16X4_F32 (wave-level 16x16 tiles, no precision change).
// ---------------------------------------------------------------------------

#define BB 64
#define II 512
#define HH 1024
#define CLIPV 2.0f
#define LN_EPSV 1e-5f
#define HCHUNKS 16
#define HPER (HH / HCHUNKS) // 64 rows of hebb per block in streaming passes

typedef float v2f __attribute__((ext_vector_type(2)));
typedef float v4f __attribute__((ext_vector_type(4))); // native vec for NT ld/st
typedef float v8f __attribute__((ext_vector_type(8)));

// ---------------------------------------------------------------------------
// Pass 1: partial[chunk][b][k] = sum_{h in chunk} h_pre[b,h]*alpha[h,k]*hebb[b,h,k]
// 1024 blocks x 256 threads, float4 per thread -> 4KB coalesced per row.
// alpha (4MB) is L2-resident across the 64 batches.
// ---------------------------------------------------------------------------
__global__ void rec_partial_kernel(const float* __restrict__ h_pre,
                                   const float* __restrict__ alpha,
                                   const float* __restrict__ hebb,
                                   float* __restrict__ partial) {
    const int b = blockIdx.x;
    const int chunk = blockIdx.y;
    const int k = threadIdx.x * 4;
    const int h0 = chunk * HPER;
    const float* hebb_b = hebb + (size_t)b * HH * HH;
    float4 acc = make_float4(0.f, 0.f, 0.f, 0.f);
    for (int hh = 0; hh < HPER; ++hh) {
        const int h = h0 + hh;
        const float hp = h_pre[b * HH + h];
        const float4 al = *(const float4*)(alpha + (size_t)h * HH + k);
        const float4 hbv = *(const float4*)(hebb_b + (size_t)h * HH + k);
        acc.x = fmaf(hp * al.x, hbv.x, acc.x);
        acc.y = fmaf(hp * al.y, hbv.y, acc.y);
        acc.z = fmaf(hp * al.z, hbv.z, acc.z);
        acc.w = fmaf(hp * al.w, hbv.w, acc.w);
    }
    *(float4*)(partial + ((size_t)chunk * BB + b) * HH + k) = acc;
}

// ---------------------------------------------------------------------------
// rec[b,k] = fc_b[k] + sum_chunk partial[chunk][b][k]
// ---------------------------------------------------------------------------
__global__ void rec_reduce_kernel(const float* __restrict__ partial,
                                  const float* __restrict__ fc_b,
                                  float* __restrict__ rec) {
    const int b = blockIdx.x;
    const int k = threadIdx.x * 4;
    float4 acc = *(const float4*)(fc_b + k);
    for (int c = 0; c < HCHUNKS; ++c) {
        const float4 p = *(const float4*)(partial + ((size_t)c * BB + b) * HH + k);
        acc.x += p.x; acc.y += p.y; acc.z += p.z; acc.w += p.w;
    }
    *(float4*)(rec + (size_t)b * HH + k) = acc;
}

// ---------------------------------------------------------------------------
// pre_gemm[m,n] = sum_k x[m,k]*fc_w[n,k]  +  sum_k h_pre[m,k]*weight[k,n]
// Exact f32 via V_WMMA_F32_16X16X4_F32. One wave owns a 16x16 tile.
// A fragment (16x4): lane L holds M=L&15, K pair kh..kh+1 where kh=(L>=16)?2:0
// B fragment (4x16): lane L holds N=L&15, rows kh..kh+1 (same pairing)
//   -> both fragments are contiguous float2 loads (B^T is fc_w row-major).
// C/D (16x16): VGPR r -> M = r + (L>=16 ? 8 : 0), N = L&15.
// ---------------------------------------------------------------------------
__global__ void pre_gemm_wmma_kernel(const float* __restrict__ x,      // [64,512]
                                     const float* __restrict__ fc_w,   // [1024,512]
                                     const float* __restrict__ h_pre,  // [64,1024]
                                     const float* __restrict__ weight, // [1024,1024]
                                     float* __restrict__ pre_gemm) {   // [64,1024]
    const int lane = threadIdx.x & 31;
    const int wave = threadIdx.x >> 5;
    const int n0 = (blockIdx.x * 4 + wave) * 16; // gridDim.x = 16
    const int m0 = blockIdx.y * 16;              // gridDim.y = 4
    const int mrow = m0 + (lane & 15);
    const int ncol = n0 + (lane & 15);
    const int kh = (lane >> 4) << 1; // 0 or 2

    v8f c = {0.f, 0.f, 0.f, 0.f, 0.f, 0.f, 0.f, 0.f};

    // GEMM 1: x @ fc_w^T  (K = 512).  B[k][n] = fc_w[n][k] -> contiguous in k.
    const float* xrow = x + (size_t)mrow * II;
    const float* wrow = fc_w + (size_t)ncol * II;
#pragma unroll 4
    for (int k0 = 0; k0 < II; k0 += 4) {
        v2f a, bb;
        a.x = xrow[k0 + kh];
        a.y = xrow[k0 + kh + 1];
        bb.x = wrow[k0 + kh];
        bb.y = wrow[k0 + kh + 1];
        c = __builtin_amdgcn_wmma_f32_16x16x4_f32(false, a, false, bb,
                                                  (short)0, c, false, false);
    }
    // GEMM 2: h_pre @ weight  (K = 1024).  B[k][n] = weight[k][n].
    const float* hrow = h_pre + (size_t)mrow * HH;
#pragma unroll 4
    for (int k0 = 0; k0 < HH; k0 += 4) {
        v2f a, bb;
        a.x = hrow[k0 + kh];
        a.y = hrow[k0 + kh + 1];
        bb.x = weight[(size_t)(k0 + kh) * HH + ncol];
        bb.y = weight[(size_t)(k0 + kh + 1) * HH + ncol];
        c = __builtin_amdgcn_wmma_f32_16x16x4_f32(false, a, false, bb,
                                                  (short)0, c, false, false);
    }
    const int mbase = m0 + ((lane >> 4) << 3);
#pragma unroll
    for (int r = 0; r < 8; ++r)
        pre_gemm[(size_t)(mbase + r) * HH + ncol] = c[r];
}

// ---------------------------------------------------------------------------
// Per batch row: pre = pre_gemm + rec; LayerNorm; tanh -> h_post;
// m = tanh(h_post . mod_w + mod_b).  One block (256 thr) per row, float4/thr.
// ---------------------------------------------------------------------------
__global__ void ln_tanh_mod_kernel(const float* __restrict__ pre_gemm,
                                   const float* __restrict__ rec,
                                   const float* __restrict__ ln_g,
                                   const float* __restrict__ ln_b,
                                   const float* __restrict__ mod_w, // [1,H]
                                   const float* __restrict__ mod_b, // [1]
                                   float* __restrict__ hpost_ws,
                                   float* __restrict__ m_ws,
                                   float* __restrict__ hpost_out,
                                   float* __restrict__ m_out) {
    __shared__ float red[256];
    const int b = blockIdx.x;
    const int t = threadIdx.x;
    const int k = t * 4;

    float4 v = *(const float4*)(pre_gemm + (size_t)b * HH + k);
    const float4 r4 = *(const float4*)(rec + (size_t)b * HH + k);
    v.x += r4.x; v.y += r4.y; v.z += r4.z; v.w += r4.w;

    // mean
    red[t] = v.x + v.y + v.z + v.w;
    __syncthreads();
    for (int off = 128; off > 0; off >>= 1) {
        if (t < off) red[t] += red[t + off];
        __syncthreads();
    }
    const float mu = red[0] * (1.0f / HH);
    __syncthreads();

    // variance
    const float dx = v.x - mu, dy = v.y - mu, dz = v.z - mu, dw = v.w - mu;
    red[t] = dx * dx + dy * dy + dz * dz + dw * dw;
    __syncthreads();
    for (int off = 128; off > 0; off >>= 1) {
        if (t < off) red[t] += red[t + off];
        __syncthreads();
    }
    const float rstd = rsqrtf(red[0] * (1.0f / HH) + LN_EPSV);
    __syncthreads();

    const float4 g = *(const float4*)(ln_g + k);
    const float4 be = *(const float4*)(ln_b + k);
    float4 hp;
    hp.x = tanhf(dx * rstd * g.x + be.x);
    hp.y = tanhf(dy * rstd * g.y + be.y);
    hp.z = tanhf(dz * rstd * g.z + be.z);
    hp.w = tanhf(dw * rstd * g.w + be.w);
    *(float4*)(hpost_ws + (size_t)b * HH + k) = hp;
    *(float4*)(hpost_out + (size_t)b * HH + k) = hp;

    // m = tanh(h_post . mod_w + mod_b)
    const float4 mw = *(const float4*)(mod_w + k);
    red[t] = hp.x * mw.x + hp.y * mw.y + hp.z * mw.z + hp.w * mw.w;
    __syncthreads();
    for (int off = 128; off > 0; off >>= 1) {
        if (t < off) red[t] += red[t + off];
        __syncthreads();
    }
    if (t == 0) {
        const float m = tanhf(red[0] + mod_b[0]);
        m_ws[b] = m;
        m_out[b] = m;
    }
}

// ---------------------------------------------------------------------------
// Pass 2: hebb_new = clip(hebb + (m[b]*modf_w[k]+modf_b[k])*h_pre[b,h]*h_post[b,k])
// hebb read is last-use, hebb_new is never re-read -> NT load/store (native
// ext_vector v4f: clang's nontemporal builtins reject HIP_vector_type) keeps
// L2 free; pass-1 RT loads left most of hebb warm in the 192MB L2.
// ---------------------------------------------------------------------------
__global__ void hebb_update_kernel(const float* __restrict__ hebb,
                                   const float* __restrict__ h_pre,
                                   const float* __restrict__ hpost,
                                   const float* __restrict__ m_ws,
                                   const float* __restrict__ modf_w, // [H,1]
                                   const float* __restrict__ modf_b, // [H]
                                   float* __restrict__ hebb_out) {
    const int b = blockIdx.x;
    const int chunk = blockIdx.y;
    const int k = threadIdx.x * 4;
    const float m = m_ws[b];
    const v4f mw = *(const v4f*)(modf_w + k);
    const v4f mb = *(const v4f*)(modf_b + k);
    const v4f hp = *(const v4f*)(hpost + (size_t)b * HH + k);
    const v4f g = (m * mw + mb) * hp; // eta[b,k] * h_post[b,k]

    const size_t base = (size_t)b * HH * HH;
    const int h0 = chunk * HPER;
    for (int hh = 0; hh < HPER; ++hh) {
        const int h = h0 + hh;
        const float pre = h_pre[b * HH + h];
        const v4f hb =
            __builtin_nontemporal_load((const v4f*)(hebb + base + (size_t)h * HH + k));
        v4f o;
        o.x = fminf(CLIPV, fmaxf(-CLIPV, fmaf(pre, g.x, hb.x)));
        o.y = fminf(CLIPV, fmaxf(-CLIPV, fmaf(pre, g.y, hb.y)));
        o.z = fminf(CLIPV, fmaxf(-CLIPV, fmaf(pre, g.z, hb.z)));
        o.w = fminf(CLIPV, fmaxf(-CLIPV, fmaf(pre, g.w, hb.w)));
        __builtin_nontemporal_store(o, (v4f*)(hebb_out + base + (size_t)h * HH + k));
    }
}

// ---------------------------------------------------------------------------
extern "C" void kernel_launch(void* const* d_in, const int* in_sizes, int n_in,
                              void* d_out, int out_size, void* d_ws, size_t ws_size,
                              hipStream_t stream) {
    const float* x      = (const float*)d_in[0];
    const float* h_pre  = (const float*)d_in[1];
    const float* hebb   = (const float*)d_in[2];
    const float* fc_w   = (const float*)d_in[3];
    const float* fc_b   = (const float*)d_in[4];
    const float* weight = (const float*)d_in[5];
    const float* alpha  = (const float*)d_in[6];
    const float* ln_g   = (const float*)d_in[7];
    const float* ln_b   = (const float*)d_in[8];
    const float* mod_w  = (const float*)d_in[9];
    const float* mod_b  = (const float*)d_in[10];
    const float* modf_w = (const float*)d_in[11];
    const float* modf_b = (const float*)d_in[12];

    // outputs: h_post [64,1024] | m [64] | hebb_new [64,1024,1024]
    float* out       = (float*)d_out;
    float* hpost_out = out;
    float* m_out     = out + BB * HH;
    float* hebb_out  = out + BB * HH + BB; // offset 65600 floats, 16B aligned

    // workspace (~5 MB)
    float* ws       = (float*)d_ws;
    float* partial  = ws;                                   // 16*64*1024
    float* rec      = partial + (size_t)HCHUNKS * BB * HH;  // 64*1024
    float* pre_gemm = rec + BB * HH;                        // 64*1024
    float* hpost_ws = pre_gemm + BB * HH;                   // 64*1024
    float* m_ws     = hpost_ws + BB * HH;                   // 64

    rec_partial_kernel<<<dim3(BB, HCHUNKS), 256, 0, stream>>>(h_pre, alpha, hebb, partial);
    rec_reduce_kernel<<<BB, 256, 0, stream>>>(partial, fc_b, rec);
    pre_gemm_wmma_kernel<<<dim3(HH / 64, BB / 16), 128, 0, stream>>>(x, fc_w, h_pre, weight, pre_gemm);
    ln_tanh_mod_kernel<<<BB, 256, 0, stream>>>(pre_gemm, rec, ln_g, ln_b, mod_w, mod_b,
                                               hpost_ws, m_ws, hpost_out, m_out);
    hebb_update_kernel<<<dim3(BB, HCHUNKS), 256, 0, stream>>>(hebb, h_pre, hpost_ws, m_ws,
                                                              modf_w, modf_b, hebb_out);
}